// GMM_Head_55095840473822
// MI455X (gfx1250) — compile-verified
//
#include <hip/hip_runtime.h>
#include <math.h>

#define INPUT_DIM   4096
#define OUTPUT_DIM  1024
#define NUM_G       128
#define BATCH       2048

typedef float v2f __attribute__((ext_vector_type(2)));
typedef float v4f __attribute__((ext_vector_type(4)));
typedef float v8f __attribute__((ext_vector_type(8)));

// ---------------------------------------------------------------------------
// Stage 1: means[b][g] = tanh(x[b,:] . W[g,:] + bias[g])   (FP32-exact WMMA)
//
// V_WMMA_F32_16X16X4_F32 fragment layout (wave32):
//   A (16x4): lane L, vgpr j -> A[M=L&15][k-slot = 2*(L>>4)+j]
//   B (4x16): identical indexing for B[k][n] = W[n][k].
// The k-slot -> actual-k mapping is ours to choose (contraction order is
// free), so each lane loads ONE b128 at  row*4096 + 8*chunk + 4*half :
//   half=0 lanes carry actual k {8c..8c+3}, half=1 lanes {8c+4..8c+7};
//   (v0,v1) feeds WMMA step A, (v2,v3) feeds step B. Same addressing for A
//   and B fragments keeps slots paired on identical actual k. One b128 per
//   matrix per 2 WMMA steps (half the load instructions of the b64 scheme).
// Loop is software-pipelined: next chunk's 3 loads issue before the current
// chunk's 4 WMMAs, hiding L2 latency at ~1 wave/SIMD occupancy.
//
// C/D (16x16 f32): lane L, vgpr r -> C[M = r + 8*(L>>4)][N = L&15]
// Grid: BATCH/16 blocks x 128 threads (4 waves); wave w = one 16-row M-tile
// vs N columns [32w, 32w+32) (two accumulators, A reused 2x).
// ---------------------------------------------------------------------------
__global__ __launch_bounds__(128) void means_kernel(
    const float* __restrict__ X, const float* __restrict__ Wg,
    const float* __restrict__ bias, float* __restrict__ means) {
  const int lane = threadIdx.x & 31;
  const int wv   = threadIdx.x >> 5;      // 0..3
  const int half = lane >> 4;             // 0/1
  const int l16  = lane & 15;
  const int mt   = blockIdx.x * 16;       // batch tile base
  const int nt0  = wv * 32;               // first gaussian tile
  const int nt1  = nt0 + 16;              // second gaussian tile

  const float* ar  = X  + (size_t)(mt  + l16) * INPUT_DIM + 4 * half;
  const float* br0 = Wg + (size_t)(nt0 + l16) * INPUT_DIM + 4 * half;
  const float* br1 = Wg + (size_t)(nt1 + l16) * INPUT_DIM + 4 * half;

  v8f acc0 = {};
  v8f acc1 = {};

  // prologue: first chunk
  v4f a  = *(const v4f*)(ar);
  v4f f0 = *(const v4f*)(br0);
  v4f f1 = *(const v4f*)(br1);

#pragma unroll 4
  for (int k = 8; k < INPUT_DIM; k += 8) {
    // prefetch next chunk before consuming current one
    v4f an = *(const v4f*)(ar  + k);
    v4f g0 = *(const v4f*)(br0 + k);
    v4f g1 = *(const v4f*)(br1 + k);

    v2f alo = {a.x, a.y},  ahi = {a.z, a.w};
    v2f l0  = {f0.x, f0.y}, h0 = {f0.z, f0.w};
    v2f l1  = {f1.x, f1.y}, h1 = {f1.z, f1.w};
    // 8 args: (neg_a, A, neg_b, B, c_mod, C, reuse_a, reuse_b)
    acc0 = __builtin_amdgcn_wmma_f32_16x16x4_f32(
        false, alo, false, l0, (short)0, acc0, false, false);
    acc1 = __builtin_amdgcn_wmma_f32_16x16x4_f32(
        false, alo, false, l1, (short)0, acc1, false, false);
    acc0 = __builtin_amdgcn_wmma_f32_16x16x4_f32(
        false, ahi, false, h0, (short)0, acc0, false, false);
    acc1 = __builtin_amdgcn_wmma_f32_16x16x4_f32(
        false, ahi, false, h1, (short)0, acc1, false, false);

    a = an; f0 = g0; f1 = g1;
  }
  // epilogue: last chunk
  {
    v2f alo = {a.x, a.y},  ahi = {a.z, a.w};
    v2f l0  = {f0.x, f0.y}, h0 = {f0.z, f0.w};
    v2f l1  = {f1.x, f1.y}, h1 = {f1.z, f1.w};
    acc0 = __builtin_amdgcn_wmma_f32_16x16x4_f32(
        false, alo, false, l0, (short)0, acc0, false, false);
    acc1 = __builtin_amdgcn_wmma_f32_16x16x4_f32(
        false, alo, false, l1, (short)0, acc1, false, false);
    acc0 = __builtin_amdgcn_wmma_f32_16x16x4_f32(
        false, ahi, false, h0, (short)0, acc0, false, false);
    acc1 = __builtin_amdgcn_wmma_f32_16x16x4_f32(
        false, ahi, false, h1, (short)0, acc1, false, false);
  }

  const float bb0 = bias[nt0 + l16];
  const float bb1 = bias[nt1 + l16];
#pragma unroll
  for (int r = 0; r < 8; ++r) {
    const int m = mt + r + 8 * half;
    means[m * NUM_G + nt0 + l16] = tanhf(acc0[r] + bb0);
    means[m * NUM_G + nt1 + l16] = tanhf(acc1[r] + bb1);
  }
}

// ---------------------------------------------------------------------------
// Stage 2: out[b][d] = log( mean_g coeff*exp(-(ep[d]-m[b][g])^2 * 131072) + 1e-5 )
//
// sigma = 2/1024 = the eval-grid spacing: a gaussian's contribution at
// the k-th nearest grid point is coeff*exp(-k^2/2) -> below 1e-25 of the 1e-5
// floor beyond k=11. One block per batch row; each of 128 threads owns one
// gaussian and scatter-adds its 23-point window into an LDS accumulator with
// ds_add_f32 atomics (~6M exps total vs 268M dense).
// ---------------------------------------------------------------------------
__global__ __launch_bounds__(128) void mixture_kernel(
    const float* __restrict__ means, float* __restrict__ out) {
  __shared__ float mix[OUTPUT_DIM];
  const int tid = threadIdx.x;            // 0..127 -> gaussian id
  const int row = blockIdx.x;             // batch row

#pragma unroll
  for (int i = tid; i < OUTPUT_DIM; i += 128) mix[i] = 0.0f;
  __syncthreads();

  const float m  = means[row * NUM_G + tid];
  // continuous grid position of the mean: ep[d] = -1 + (2d+1)/1024
  const float cf = (m + 1.0f) * 512.0f - 0.5f;
  const int   c  = (int)floorf(cf + 0.5f);
  const int   lo = max(0, c - 11);
  const int   hi = min(OUTPUT_DIM - 1, c + 11);
  const float coeff = 204.2584477f;       // 512 / sqrt(2*pi)

  for (int d = lo; d <= hi; ++d) {
    const float ep   = -1.0f + (2.0f * (float)d + 1.0f) * (1.0f / 1024.0f);
    const float diff = ep - m;
    const float g    = coeff * __expf(-131072.0f * diff * diff); // 1/(2s^2)
    atomicAdd(&mix[d], g);                // -> ds_add_f32 (no return)
  }
  __syncthreads();

  const float inv_g = 1.0f / (float)NUM_G;
#pragma unroll
  for (int i = tid; i < OUTPUT_DIM; i += 128) {
    out[(size_t)row * OUTPUT_DIM + i] = logf(mix[i] * inv_g + 1e-5f);
  }
}

// ---------------------------------------------------------------------------
extern "C" void kernel_launch(void* const* d_in, const int* in_sizes, int n_in,
                              void* d_out, int out_size, void* d_ws, size_t ws_size,
                              hipStream_t stream) {
  const float* x  = (const float*)d_in[0];   // (2048, 4096)
  const float* W  = (const float*)d_in[1];   // (128, 4096)
  const float* b  = (const float*)d_in[2];   // (128,)
  float* out   = (float*)d_out;              // (2048, 1024)
  float* means = (float*)d_ws;               // (2048, 128) scratch, 1 MB

  means_kernel<<<dim3(BATCH / 16), dim3(128), 0, stream>>>(x, W, b, means);
  mixture_kernel<<<dim3(BATCH), dim3(128), 0, stream>>>(means, out);
}